// IFNode_25838523253092
// MI455X (gfx1250) — compile-verified
//
#include <hip/hip_runtime.h>

// IF-neuron forward scan over T=16 time steps.
// x: (T, B*C*H*W) fp32 -> spikes: same shape.
// Memory-bound (AI ~0.38 flop/byte, 512 MB traffic, ~22 us @ 23.3 TB/s):
// stream x once via CDNA5 async global->LDS 16-deep ASYNCcnt pipeline,
// carry membrane potential v in VGPRs across all T, non-temporal b128
// stores for the spike output. Both streams > 192 MB L2 => NT both ways.

#define TT   16     // time steps (reference: T = 16)
#define BLK  256    // 8 waves of 32

// CPol temporal hint: TH[2:0] = 1 -> TH_LOAD_NT (read-once stream).
#define CPOL_LOAD_NT 1

typedef float v4f __attribute__((ext_vector_type(4)));
typedef int   v4i __attribute__((ext_vector_type(4)));

// Pointer types matching the async-LDS builtin: 16-byte vector pointee,
// global (AS1) source and LDS (AS3) destination.
typedef __attribute__((address_space(1))) v4i* as1_v4i_p;
typedef __attribute__((address_space(3))) v4i* as3_v4i_p;

// One 16-byte async copy global -> LDS (per lane). Tracked by ASYNCcnt.
__device__ __forceinline__ void async_ld16(const void* g, void* l) {
  __builtin_amdgcn_global_load_async_to_lds_b128(
      (as1_v4i_p)g, (as3_v4i_p)l, /*imm offset*/ 0, /*cpol*/ CPOL_LOAD_NT);
}

// Drain the async pipeline: step t waits for ASYNCcnt <= (TT-1-t), i.e. the
// t-th issued load (in-order completion) has landed in LDS, then integrates.
// Template recursion so the wait count is a source-level immediate.
template <int t>
__device__ __forceinline__ void if_steps(const v4f* lds, int tid,
                                         float* outp, long nflt, v4f& v) {
  if constexpr (t < TT) {
    __builtin_amdgcn_s_wait_asynccnt((unsigned short)(TT - 1 - t));
    __asm__ __volatile__("" ::: "memory");   // keep ds_load below the wait

    v4f x = lds[t * BLK + tid];              // ds_load_b128 (lane-private slot)
    v4f h = v + x;
    v4f s, vn;
#pragma unroll
    for (int k = 0; k < 4; ++k) {
      float hk    = h[k];
      bool  spike = (hk >= 1.0f);            // v_threshold = 1.0
      s[k]  = spike ? 1.0f : 0.0f;           // heaviside forward
      vn[k] = spike ? 0.0f : hk;             // hard reset to v_reset = 0
    }
    v = vn;
    // Streaming output: non-temporal 128-bit store (don't pollute L2).
    __builtin_nontemporal_store(s, (v4f*)(outp + (long)t * nflt));

    if_steps<t + 1>(lds, tid, outp, nflt, v);
  }
}

__global__ __launch_bounds__(BLK)
void IFNode_25838523253092_kernel(const float* __restrict__ x,
                                  float* __restrict__ out, int n4) {
  // 64 KB LDS staging: TT slices x (BLK lanes x 16 B). 5 blocks/WGP of 320 KB.
  __shared__ v4f lds[TT * BLK];

  const int  tid  = threadIdx.x;
  const long idx4 = (long)blockIdx.x * BLK + tid;   // float4 index in a slice
  if (idx4 >= n4) return;

  const long   nflt = (long)n4 * 4;                 // floats per time slice
  const float* gx   = x + idx4 * 4;

  // Issue all TT async slice loads up front: 16-deep pipeline per wave,
  // 8 KB in flight per wave; loads for later t overlap compute of earlier t.
#pragma unroll
  for (int t = 0; t < TT; ++t) {
    async_ld16(gx + (long)t * nflt, &lds[t * BLK + tid]);
  }

  v4f v = {0.0f, 0.0f, 0.0f, 0.0f};                 // v[-1] = 0
  float* go = out + idx4 * 4;
  if_steps<0>(lds, tid, go, nflt, v);
}

extern "C" void kernel_launch(void* const* d_in, const int* in_sizes, int n_in,
                              void* d_out, int out_size, void* d_ws, size_t ws_size,
                              hipStream_t stream) {
  (void)n_in; (void)d_ws; (void)ws_size; (void)out_size;
  const float* x   = (const float*)d_in[0];
  float*       out = (float*)d_out;

  const long total  = (long)in_sizes[0];            // T*B*C*H*W = 67,108,864
  const int  n4     = (int)(total / (TT * 4));      // float4 cols per slice
  const int  blocks = (n4 + BLK - 1) / BLK;         // 4096 blocks

  IFNode_25838523253092_kernel<<<blocks, BLK, 0, stream>>>(x, out, n4);
}